// OTLoss_68341519614325
// MI455X (gfx1250) — compile-verified
//
#include <hip/hip_runtime.h>

// ---- problem constants (match reference) ----
#define NPTS   512          // points per sample
#define NB     64           // grid blocks per dim
#define GRID   (NB*NB)      // 4096 grid cells
#define NITER  100
#define REGC   10.0f
#define EPSC   1e-8f
#define M_EPSC 1e-16f
#define BATCH  16
#define TSTR   516          // padded row stride (floats) for KyT/KxT: bank-conflict free

typedef __attribute__((ext_vector_type(2))) float v2f;
typedef __attribute__((ext_vector_type(8))) float v8f;

__device__ __forceinline__ float red16(float v) {
    // butterfly reduce within each 16-lane half of the wave32
    v += __shfl_xor(v, 1, 32);
    v += __shfl_xor(v, 2, 32);
    v += __shfl_xor(v, 4, 32);
    v += __shfl_xor(v, 8, 32);
    return v;
}
__device__ __forceinline__ float red32(float v) {
    v += __shfl_xor(v, 1, 32);
    v += __shfl_xor(v, 2, 32);
    v += __shfl_xor(v, 4, 32);
    v += __shfl_xor(v, 8, 32);
    v += __shfl_xor(v, 16, 32);
    return v;
}

__device__ __forceinline__ v8f wmma_f32(v2f a, v2f b, v8f c) {
    return __builtin_amdgcn_wmma_f32_16x16x4_f32(
        /*neg_a=*/false, a, /*neg_b=*/false, b,
        /*c_mod=*/(short)0, c, /*reuse_a=*/false, /*reuse_b=*/false);
}

__global__ void zero_out3(float* o) {
    if (threadIdx.x < 3) o[threadIdx.x] = 0.0f;
}

// One workgroup per batch sample; 512 threads = 16 waves (wave32).
__global__ __launch_bounds__(512)
void ot_sinkhorn_kernel(const float* __restrict__ pred,
                        const float* __restrict__ normed,
                        const float* __restrict__ pts,
                        float* __restrict__ out) {
    extern __shared__ float smem[];
    float* KyT_s = smem;                    // [NB][TSTR]  Ky transposed: [gy][n]
    float* KxT_s = KyT_s + NB*TSTR;         // [NB][TSTR]  Kx transposed: [gx][n]
    float* V_s   = KxT_s + NB*TSTR;         // [NB][NB]    v reshaped (gy, gx); later beta
    float* nrm_s = V_s   + GRID;            // [GRID]
    float* u_s   = nrm_s + GRID;            // [NPTS]
    float* x_s   = u_s   + NPTS;            // [NPTS]
    float* y_s   = x_s   + NPTS;            // [NPTS]
    float* red_s = y_s   + NPTS;            // [8] block accumulators

    const int b    = blockIdx.x;
    const int tid  = threadIdx.x;
    const int lane = tid & 31;
    const int wave = tid >> 5;              // 0..15

    // WMMA per-lane fragment coordinates (ISA 7.12.2, wave32, f32 shapes)
    const int mA    = lane & 15;            // A row / B col
    const int kh    = (lane >> 4) << 1;     // K pair base: 0 or 2
    const int half8 = (lane >> 4) << 3;     // C row offset: 0 or 8
    const int colC  = lane & 15;            // C col

    // ---------------- setup: factorized kernels into LDS (transposed) ----------------
    if (tid < 8) red_s[tid] = 0.0f;
    {
        const int n = tid;                  // 512 threads <-> 512 points
        const float x = pts[(b*NPTS + n)*2 + 0];
        const float y = pts[(b*NPTS + n)*2 + 1];
        x_s[n] = x; y_s[n] = y;
        u_s[n] = 1.0f / (float)NPTS;
        #pragma unroll 8
        for (int g = 0; g < NB; ++g) {
            const float c  = (float)g * 8.0f + 4.0f;
            const float dx = x - c, dy = y - c;
            KxT_s[g*TSTR + n] = expf(-dx*dx * (1.0f/REGC));
            KyT_s[g*TSTR + n] = expf(-dy*dy * (1.0f/REGC));
        }
    }
    for (int g = tid; g < GRID; g += 512) nrm_s[g] = normed[b*GRID + g];
    __syncthreads();

    // ---------------- Sinkhorn iterations ----------------
    for (int it = 0; it < NITER; ++it) {
        // GEMM1: KTu[gy,gx] = sum_n (u[n]*Ky[n,gy]) * Kx[n,gx]
        // one 16x16 tile per wave; A/B/u pairs are contiguous b64 loads; pipelined.
        {
            const int gyT = (wave >> 2) << 4;
            const int gxT = (wave &  3) << 4;
            const float* Ap = &KyT_s[(gyT + mA)   * TSTR];
            const float* Bp = &KxT_s[(gxT + colC) * TSTR];
            v2f a  = *(const v2f*)&Ap[kh];
            v2f uu = *(const v2f*)&u_s[kh];
            v2f bb = *(const v2f*)&Bp[kh];
            v8f c = {};
            for (int n0 = 4; n0 < NPTS; n0 += 4) {
                const v2f an = *(const v2f*)&Ap[n0 + kh];     // prefetch k+1
                const v2f un = *(const v2f*)&u_s[n0 + kh];
                const v2f bn = *(const v2f*)&Bp[n0 + kh];
                c = wmma_f32(uu * a, bb, c);                  // consume k
                a = an; uu = un; bb = bn;
            }
            c = wmma_f32(uu * a, bb, c);
            #pragma unroll
            for (int r = 0; r < 8; ++r) {
                const int gy = gyT + r + half8;
                const int gx = gxT + colC;
                V_s[gy*NB + gx] = nrm_s[gy*NB + gx] / (c[r] + M_EPSC);
            }
        }
        __syncthreads();

        // GEMM2: M[n,gx] = sum_gy Ky[n,gy]*V[gy,gx]; Kv[n] = sum_gx Kx[n,gx]*M[n,gx]
        // 2 n-blocks per wave; A pair shared across 4 gx-tiles; pipelined.
        for (int blk = 0; blk < 2; ++blk) {
            const int nb = ((wave << 1) + blk) << 4;
            v8f acc[4] = {{}, {}, {}, {}};
            v2f a, bb[4];
            a.x = KyT_s[(kh    )*TSTR + nb + mA];
            a.y = KyT_s[(kh + 1)*TSTR + nb + mA];
            #pragma unroll
            for (int gxt = 0; gxt < 4; ++gxt) {
                bb[gxt].x = V_s[(kh    )*NB + (gxt<<4) + colC];
                bb[gxt].y = V_s[(kh + 1)*NB + (gxt<<4) + colC];
            }
            for (int gy0 = 4; gy0 < NB; gy0 += 4) {
                const int gk = gy0 + kh;
                v2f an, bn[4];
                an.x = KyT_s[(gk    )*TSTR + nb + mA];         // prefetch k+1
                an.y = KyT_s[(gk + 1)*TSTR + nb + mA];
                #pragma unroll
                for (int gxt = 0; gxt < 4; ++gxt) {
                    bn[gxt].x = V_s[(gk    )*NB + (gxt<<4) + colC];
                    bn[gxt].y = V_s[(gk + 1)*NB + (gxt<<4) + colC];
                }
                #pragma unroll
                for (int gxt = 0; gxt < 4; ++gxt)              // consume k
                    acc[gxt] = wmma_f32(a, bb[gxt], acc[gxt]);
                a = an;
                #pragma unroll
                for (int gxt = 0; gxt < 4; ++gxt) bb[gxt] = bn[gxt];
            }
            #pragma unroll
            for (int gxt = 0; gxt < 4; ++gxt)
                acc[gxt] = wmma_f32(a, bb[gxt], acc[gxt]);

            // fused rowsum with Kx + u update
            #pragma unroll
            for (int r = 0; r < 8; ++r) {
                const int n = nb + r + half8;
                float kvr = KxT_s[(0*16 + colC)*TSTR + n] * acc[0][r]
                          + KxT_s[(1*16 + colC)*TSTR + n] * acc[1][r]
                          + KxT_s[(2*16 + colC)*TSTR + n] * acc[2][r]
                          + KxT_s[(3*16 + colC)*TSTR + n] * acc[3][r];
                kvr = red16(kvr);                             // sum over 16 cols
                if ((lane & 15) == 0)
                    u_s[n] = (1.0f/(float)NPTS) / (kvr + M_EPSC);
            }
        }
        __syncthreads();
    }

    // ---------------- epilogue ----------------
    // red_s: [0]=ot_obj, [1]=sc, [2]=sum(pred*beta), [3]=loss, [4]=w_dist

    // w_dist = sum_n u[n] * [ Kx . (KyD @ V) + (xd*Kx) . (Ky @ V) ]  (fp32 WMMA)
    // Must run before V_s is overwritten with beta.
    {
        float s_w = 0.0f;
        for (int blk = 0; blk < 2; ++blk) {
            const int nb = ((wave << 1) + blk) << 4;
            const float yA = y_s[nb + mA];
            for (int gxt = 0; gxt < 4; ++gxt) {
                v8f c1 = {};   // with A = yd*Ky
                v8f c2 = {};   // with A = Ky
                for (int gy0 = 0; gy0 < NB; gy0 += 4) {
                    const int gk = gy0 + kh;
                    const float ky0 = KyT_s[(gk    )*TSTR + nb + mA];
                    const float ky1 = KyT_s[(gk + 1)*TSTR + nb + mA];
                    const float cy0 = (float)gk * 8.0f + 4.0f;
                    const float cy1 = cy0 + 8.0f;
                    const float yd0 = (yA - cy0)*(yA - cy0);
                    const float yd1 = (yA - cy1)*(yA - cy1);
                    v2f a1, a2, bb;
                    a1.x = yd0*ky0; a1.y = yd1*ky1;
                    a2.x = ky0;     a2.y = ky1;
                    bb.x = V_s[(gk    )*NB + (gxt<<4) + colC];
                    bb.y = V_s[(gk + 1)*NB + (gxt<<4) + colC];
                    c1 = wmma_f32(a1, bb, c1);
                    c2 = wmma_f32(a2, bb, c2);
                }
                #pragma unroll
                for (int r = 0; r < 8; ++r) {
                    const int n  = nb + r + half8;
                    const int gx = (gxt<<4) + colC;
                    const float cx = (float)gx * 8.0f + 4.0f;
                    float xd = x_s[n] - cx; xd *= xd;
                    const float kx = KxT_s[gx*TSTR + n];
                    s_w += u_s[n] * (kx*c1[r] + xd*kx*c2[r]);
                }
            }
        }
        s_w = red32(s_w);
        if (lane == 0) atomicAdd(&red_s[4], s_w);
    }
    __syncthreads();

    // pass 1: beta = REG*log(v); overwrite V_s with beta; reduce ot_obj, sc, sum(pred*beta)
    {
        float s_ot = 0.0f, s_sc = 0.0f, s_pb = 0.0f;
        for (int g = tid; g < GRID; g += 512) {
            const float beta = REGC * logf(V_s[g]);
            V_s[g] = beta;
            const float p = pred[b*GRID + g];
            s_ot += nrm_s[g] * beta;
            s_sc += p;
            s_pb += p * beta;
        }
        s_ot = red32(s_ot); s_sc = red32(s_sc); s_pb = red32(s_pb);
        if (lane == 0) {
            atomicAdd(&red_s[0], s_ot);
            atomicAdd(&red_s[1], s_sc);
            atomicAdd(&red_s[2], s_pb);
        }
    }
    __syncthreads();

    // pass 2: loss = sum(pred * grad)
    {
        const float sc    = red_s[1];
        const float S_pb  = red_s[2];
        const float denom = sc*sc + EPSC;
        float s_loss = 0.0f;
        for (int g = tid; g < GRID; g += 512) {
            const float beta = V_s[g];
            const float grad = (sc/denom)*beta - S_pb/denom;
            s_loss += pred[b*GRID + g] * grad;
        }
        s_loss = red32(s_loss);
        if (lane == 0) atomicAdd(&red_s[3], s_loss);
    }
    __syncthreads();

    if (tid == 0) {
        atomicAdd(&out[0], red_s[3]);   // loss
        atomicAdd(&out[1], red_s[4]);   // w_dist
        atomicAdd(&out[2], red_s[0]);   // ot_obj
    }
}

extern "C" void kernel_launch(void* const* d_in, const int* in_sizes, int n_in,
                              void* d_out, int out_size, void* d_ws, size_t ws_size,
                              hipStream_t stream) {
    const float* pred   = (const float*)d_in[0];   // (16,1,64,64)
    const float* normed = (const float*)d_in[1];   // (16,1,64,64)
    const float* pts    = (const float*)d_in[2];   // (16,512,2)
    float* out = (float*)d_out;                    // [loss, w_dist, ot_obj]

    const size_t shmem_bytes =
        (size_t)(2*NB*TSTR + 2*GRID + 3*NPTS + 8) * sizeof(float); // ~303 KB (<320 KB/WGP)

    (void)hipFuncSetAttribute((const void*)ot_sinkhorn_kernel,
                              hipFuncAttributeMaxDynamicSharedMemorySize,
                              (int)shmem_bytes);

    zero_out3<<<1, 32, 0, stream>>>(out);
    ot_sinkhorn_kernel<<<BATCH, 512, shmem_bytes, stream>>>(pred, normed, pts, out);
}